// Taylor_71949292142868
// MI455X (gfx1250) — compile-verified
//
#include <hip/hip_runtime.h>
#include <hip/hip_bf16.h>
#include <math.h>

// Taylor subpixel refinement (grid_type='norm', flip=True).
// Shapes fixed by the reference: coords [N,P,2], heatmaps [N,P,H,W],
// N=64, P=17, H=W=192. Output [N,P,2] float32 in (y, x) normalized order.
//
// MI455X reasoning: 1088 keypoints -> 34 wave32 waves; ~57 KB of scattered
// reads; entirely launch/latency-bound (HBM time < 0.1 us at 23.3 TB/s even
// if every gather misses to a distinct 128B line). Strategy: one lane per
// keypoint, issue all 13 stencil gathers as one vmem clause (single
// s_wait_loadcnt), then the log + 2x2 Newton solve in registers.
// WMMA is deliberately not used: the reference has no matrix op — dynamic
// gathers feed a log nonlinearity and a sparse K=13 stencil; no 16x16xK
// WMMA shape applies and tiling would only add permute latency.

#define TH 192
#define TW 192

__global__ __launch_bounds__(256) void taylor_refine_kernel(
    const float* __restrict__ coords,   // [NP, 2]
    const float* __restrict__ heat,     // [NP, TH, TW]
    float* __restrict__ out,            // [NP, 2]  (y_norm, x_norm)
    int NP)
{
    int i = blockIdx.x * blockDim.x + threadIdx.x;
    if (i >= NP) return;

    // coords[i] = (cx, cy) in [0,1)
    const float2 c = reinterpret_cast<const float2*>(coords)[i];
    float x = c.x * (float)TW;          // pixel-space x
    float y = c.y * (float)TH;          // pixel-space y

    int px = (int)x;                    // truncation (== floor for x>=0)
    int py = (int)y;
    bool inb = (px > 1) & (px < TW - 2) & (py > 1) & (py < TH - 2);
    int pxc = min(max(px, 2), TW - 3);
    int pyc = min(max(py, 2), TH - 3);

    const float* b = heat + (size_t)i * (TH * TW) + (size_t)pyc * TW + pxc;

    // ---- issue all 13 gathers up front (one vmem clause, one wait) ----
    float v_0p1  = b[1];
    float v_0m1  = b[-1];
    float v_p10  = b[TW];
    float v_m10  = b[-TW];
    float v_0p2  = b[2];
    float v_00   = b[0];
    float v_0m2  = b[-2];
    float v_p1p1 = b[TW + 1];
    float v_m1p1 = b[-TW + 1];
    float v_p1m1 = b[TW - 1];
    float v_m1m1 = b[-TW - 1];
    float v_p20  = b[2 * TW];
    float v_m20  = b[-2 * TW];

    // ---- log with the reference's zero guard ----
#define LG(v) logf(((v) == 0.0f) ? 1e-10f : (v))
    float g_0p1  = LG(v_0p1);
    float g_0m1  = LG(v_0m1);
    float g_p10  = LG(v_p10);
    float g_m10  = LG(v_m10);
    float g_0p2  = LG(v_0p2);
    float g_00   = LG(v_00);
    float g_0m2  = LG(v_0m2);
    float g_p1p1 = LG(v_p1p1);
    float g_m1p1 = LG(v_m1p1);
    float g_p1m1 = LG(v_p1m1);
    float g_m1m1 = LG(v_m1m1);
    float g_p20  = LG(v_p20);
    float g_m20  = LG(v_m20);
#undef LG

    // ---- finite-difference gradient / Hessian ----
    float dx  = 0.5f  * (g_0p1 - g_0m1);
    float dy  = 0.5f  * (g_p10 - g_m10);
    float dxx = 0.25f * (g_0p2 - 2.0f * g_00 + g_0m2);
    float dxy = 0.25f * (g_p1p1 - g_m1p1 - g_p1m1 + g_m1m1);
    float dyy = 0.25f * (g_p20 - 2.0f * g_00 + g_m20);

    float det = dxx * dyy - dxy * dxy;
    bool ok = inb & (det != 0.0f);
    float sdet = ok ? det : 1.0f;

    // offset = -H^{-1} [dx, dy]
    float offx = -(dyy * dx - dxy * dy) / sdet;
    float offy = -(dxx * dy - dxy * dx) / sdet;

    if (ok) { x += offx; y += offy; }

    // normalize back with true IEEE division (matches reference xy / dims
    // bit-exactly; 1/192 is not representable so a reciprocal-multiply
    // would be 1 ulp off), then flip -> (y, x).
    float2 r;
    r.x = y / (float)TH;
    r.y = x / (float)TW;
    reinterpret_cast<float2*>(out)[i] = r;
}

extern "C" void kernel_launch(void* const* d_in, const int* in_sizes, int n_in,
                              void* d_out, int out_size, void* d_ws, size_t ws_size,
                              hipStream_t stream) {
    const float* coords = (const float*)d_in[0];   // [N,P,2] float32
    const float* heat   = (const float*)d_in[1];   // [N,P,192,192] float32
    float* out          = (float*)d_out;           // [N,P,2] float32

    int NP = in_sizes[0] / 2;                      // N*P keypoints (1088)

    const int threads = 256;                       // 8 wave32 waves / block
    int blocks = (NP + threads - 1) / threads;     // 5 blocks
    taylor_refine_kernel<<<blocks, threads, 0, stream>>>(coords, heat, out, NP);
}